// GraphSAGE_68143951118848
// MI455X (gfx1250) — compile-verified
//
#include <hip/hip_runtime.h>
#include <hip/hip_bf16.h>

#define NNODES 100000
#define NEDGES 1600000
#define KIN    128

typedef __attribute__((ext_vector_type(2))) float v2f;
typedef __attribute__((ext_vector_type(8))) float v8f;

__global__ void zero4_kernel(float4* __restrict__ p, int n4) {
    int i = blockIdx.x * blockDim.x + threadIdx.x;
    if (i < n4) p[i] = make_float4(0.f, 0.f, 0.f, 0.f);
}

__global__ void deg_kernel(const int* __restrict__ dst, float* __restrict__ deg) {
    int e = blockIdx.x * blockDim.x + threadIdx.x;
    if (e < NEDGES) atomicAdd(&deg[dst[e]], 1.0f);
}

__global__ void invdeg_kernel(const float* __restrict__ deg, float* __restrict__ invdeg) {
    int i = blockIdx.x * blockDim.x + threadIdx.x;
    if (i < NNODES) invdeg[i] = 1.0f / fmaxf(deg[i], 1.0f);
}

// One wave32 per edge: lane L gathers float4 at feature offset 4*L (coalesced
// 512B per edge, L2-resident) and scatters 4 f32 atomics into agg[dst].
__global__ void scatter_kernel(const float* __restrict__ h,
                               const int* __restrict__ src,
                               const int* __restrict__ dst,
                               float* __restrict__ agg) {
    int gid  = blockIdx.x * blockDim.x + threadIdx.x;
    int e    = gid >> 5;
    int lane = gid & 31;
    int s = src[e];
    int d = dst[e];
    const float4 v = ((const float4*)(h + (size_t)s * KIN))[lane];
    float* a = agg + (size_t)d * KIN + lane * 4;
    atomicAdd(a + 0, v.x);
    atomicAdd(a + 1, v.y);
    atomicAdd(a + 2, v.z);
    atomicAdd(a + 3, v.w);
}

// Pack W [KIN x nout] row-major into WMMA B-fragment order, padded to 16-col
// tiles with zeros. float2 element pid = tile*1024 + i*32 + lane holds
// (W[k][col], W[k+1][col]) with col = tile*16 + (lane&15), k = 4*i + 2*(lane>>4).
// GEMM inner loop then does ONE coalesced global_load_b64 per B fragment.
__global__ void pack_w_kernel(const float* __restrict__ W, float2* __restrict__ Wp,
                              int nout, int ntiles) {
    int pid = blockIdx.x * blockDim.x + threadIdx.x;
    int total = ntiles * (KIN / 4) * 32;
    if (pid >= total) return;
    int lane = pid & 31;
    int i    = (pid >> 5) & ((KIN / 4) - 1);
    int tile = pid >> 10;                    // / (32 * KIN/4)
    int l = lane & 15, half = lane >> 4;
    int col = tile * 16 + l;
    int k   = 4 * i + 2 * half;
    float x = 0.f, y = 0.f;
    if (col < nout) {
        x = W[(size_t)k       * nout + col];
        y = W[(size_t)(k + 1) * nout + col];
    }
    Wp[pid] = make_float2(x, y);
}

// out[16-node tile] = hself @ Wself + (agg * invdeg) @ Wneigh + bias, [relu], [l2norm]
// One wave per 16-column output tile; fp32 WMMA 16x16x4, K = 128 per half.
// A layout (ISA 7.12.2): lanes 0-15: M=lane, vgpr0/1 = K {k,k+1}; lanes 16-31: K {k+2,k+3}.
template<int NOUT, bool ACT, bool NORM>
__global__ void __launch_bounds__(256)
sage_gemm(const float* __restrict__ hself,
          const float* __restrict__ agg,
          const float* __restrict__ invdeg,
          const float2* __restrict__ Wsp,    // packed fragments (see pack_w_kernel)
          const float2* __restrict__ Wnp,
          const float* __restrict__ bself,
          const float* __restrict__ bneigh,
          float* __restrict__ out) {
    constexpr bool FULL = (NOUT % 16) == 0;  // no column masking needed
    const int lane = threadIdx.x & 31;
    const int wv   = threadIdx.x >> 5;
    const int l    = lane & 15;
    const int half = lane >> 4;
    const int m0   = blockIdx.x * 16;        // NNODES % 16 == 0, no row guard
    const int n0   = wv * 16;
    const int col  = n0 + l;
    const bool colok = FULL || (col < NOUT);
    const int arow = m0 + l;

    const float idg = invdeg[arow];
    const float*  __restrict__ As = hself + (size_t)arow * KIN + 2 * half;
    const float*  __restrict__ An = agg   + (size_t)arow * KIN + 2 * half;
    const float2* __restrict__ Bs = Wsp + (size_t)wv * (KIN / 4) * 32 + lane;
    const float2* __restrict__ Bn = Wnp + (size_t)wv * (KIN / 4) * 32 + lane;

    v8f acc = {0.f, 0.f, 0.f, 0.f, 0.f, 0.f, 0.f, 0.f};

    // self half: h @ Wself
    #pragma unroll
    for (int i = 0; i < KIN / 4; ++i) {
        const float2 af = *(const float2*)(As + 4 * i);
        const float2 bf = Bs[i * 32];
        v2f a = {af.x, af.y};
        v2f b = {bf.x, bf.y};
        acc = __builtin_amdgcn_wmma_f32_16x16x4_f32(false, a, false, b,
                                                    (short)0, acc, false, false);
    }
    // neighbor half: (agg/deg) @ Wneigh  (mean folded into A load)
    #pragma unroll
    for (int i = 0; i < KIN / 4; ++i) {
        const float2 af = *(const float2*)(An + 4 * i);
        const float2 bf = Bn[i * 32];
        v2f a = {af.x * idg, af.y * idg};
        v2f b = {bf.x, bf.y};
        acc = __builtin_amdgcn_wmma_f32_16x16x4_f32(false, a, false, b,
                                                    (short)0, acc, false, false);
    }

    const float bias = colok ? (bself[col] + bneigh[col]) : 0.0f;
    #pragma unroll
    for (int r = 0; r < 8; ++r) {
        float v = acc[r] + bias;
        if (ACT) v = fmaxf(v, 0.0f);
        acc[r] = v;
    }

    if (NORM) {
        // C/D layout: vgpr r, lane -> row (r + 8*half), col (n0 + l).
        // Cross-wave per-row sum of squares via LDS f32 atomics.
        __shared__ float rowsum[16];
        if (threadIdx.x < 16) rowsum[threadIdx.x] = 0.0f;
        __syncthreads();
        #pragma unroll
        for (int r = 0; r < 8; ++r)
            atomicAdd(&rowsum[r + 8 * half], acc[r] * acc[r]);
        __syncthreads();
        #pragma unroll
        for (int r = 0; r < 8; ++r) {
            float nrm = fmaxf(sqrtf(rowsum[r + 8 * half]), 1e-12f);
            acc[r] = acc[r] / nrm;
        }
    }

    if (colok) {
        #pragma unroll
        for (int r = 0; r < 8; ++r) {
            const int m = m0 + r + 8 * half;
            out[(size_t)m * NOUT + col] = acc[r];
        }
    }
}

extern "C" void kernel_launch(void* const* d_in, const int* in_sizes, int n_in,
                              void* d_out, int out_size, void* d_ws, size_t ws_size,
                              hipStream_t stream) {
    const float* feat = (const float*)d_in[0];
    const int*   src  = (const int*)d_in[1];
    const int*   dstv = (const int*)d_in[2];
    const float* Ws0  = (const float*)d_in[3];
    const float* bs0  = (const float*)d_in[4];
    const float* Wn0  = (const float*)d_in[5];
    const float* bn0  = (const float*)d_in[6];
    const float* Ws1  = (const float*)d_in[7];
    const float* bs1  = (const float*)d_in[8];
    const float* Wn1  = (const float*)d_in[9];
    const float* bn1  = (const float*)d_in[10];
    const float* Ws2  = (const float*)d_in[11];
    const float* bs2  = (const float*)d_in[12];
    const float* Wn2  = (const float*)d_in[13];
    const float* bn2  = (const float*)d_in[14];
    float* out = (float*)d_out;

    // workspace layout (~155 MB): agg | h1 | h2 | deg | invdeg | packed weights
    float* agg    = (float*)d_ws;
    float* h1     = agg + (size_t)NNODES * KIN;
    float* h2     = h1  + (size_t)NNODES * KIN;
    float* deg    = h2  + (size_t)NNODES * KIN;
    float* invdeg = deg + NNODES;
    const size_t PACK8 = (size_t)8 * (KIN / 4) * 32;   // float2 count, 8 tiles
    const size_t PACK3 = (size_t)3 * (KIN / 4) * 32;   // float2 count, 3 tiles
    float2* Wsp0 = (float2*)(invdeg + NNODES);
    float2* Wnp0 = Wsp0 + PACK8;
    float2* Wsp1 = Wnp0 + PACK8;
    float2* Wnp1 = Wsp1 + PACK8;
    float2* Wsp2 = Wnp1 + PACK8;
    float2* Wnp2 = Wsp2 + PACK3;

    const int ZB = 256;
    const int featN = NNODES * KIN;
    const int aggBlocks4 = (featN / 4 + ZB - 1) / ZB;
    const int edgeWaves  = NEDGES / 8;      // 8 edges (waves) per 256-thread block
    const int gemmBlocks = NNODES / 16;

    // pack all weight matrices into WMMA fragment order (once per launch)
    pack_w_kernel<<<(int)(PACK8 + ZB - 1) / ZB, ZB, 0, stream>>>(Ws0, Wsp0, 128, 8);
    pack_w_kernel<<<(int)(PACK8 + ZB - 1) / ZB, ZB, 0, stream>>>(Wn0, Wnp0, 128, 8);
    pack_w_kernel<<<(int)(PACK8 + ZB - 1) / ZB, ZB, 0, stream>>>(Ws1, Wsp1, 128, 8);
    pack_w_kernel<<<(int)(PACK8 + ZB - 1) / ZB, ZB, 0, stream>>>(Wn1, Wnp1, 128, 8);
    pack_w_kernel<<<(int)(PACK3 + ZB - 1) / ZB, ZB, 0, stream>>>(Ws2, Wsp2, 40, 3);
    pack_w_kernel<<<(int)(PACK3 + ZB - 1) / ZB, ZB, 0, stream>>>(Wn2, Wnp2, 40, 3);

    // degrees (recomputed every call: deterministic, graph-capture safe)
    zero4_kernel<<<(NNODES / 4 + ZB - 1) / ZB, ZB, 0, stream>>>((float4*)deg, NNODES / 4);
    deg_kernel<<<(NEDGES + ZB - 1) / ZB, ZB, 0, stream>>>(dstv, deg);
    invdeg_kernel<<<(NNODES + ZB - 1) / ZB, ZB, 0, stream>>>(deg, invdeg);

    // layer 0: in(128) -> hidden(128), relu + l2norm
    zero4_kernel<<<aggBlocks4, ZB, 0, stream>>>((float4*)agg, featN / 4);
    scatter_kernel<<<edgeWaves, 256, 0, stream>>>(feat, src, dstv, agg);
    sage_gemm<128, true, true><<<gemmBlocks, 256, 0, stream>>>(
        feat, agg, invdeg, Wsp0, Wnp0, bs0, bn0, h1);

    // layer 1: hidden -> hidden, relu + l2norm
    zero4_kernel<<<aggBlocks4, ZB, 0, stream>>>((float4*)agg, featN / 4);
    scatter_kernel<<<edgeWaves, 256, 0, stream>>>(h1, src, dstv, agg);
    sage_gemm<128, true, true><<<gemmBlocks, 256, 0, stream>>>(
        h1, agg, invdeg, Wsp1, Wnp1, bs1, bn1, h2);

    // layer 2: hidden -> classes(40), plain
    zero4_kernel<<<aggBlocks4, ZB, 0, stream>>>((float4*)agg, featN / 4);
    scatter_kernel<<<edgeWaves, 256, 0, stream>>>(h2, src, dstv, agg);
    sage_gemm<40, false, false><<<gemmBlocks, 96, 0, stream>>>(
        h2, agg, invdeg, Wsp2, Wnp2, bs2, bn2, out);
}